// LengthAdaptor_55705725829267
// MI455X (gfx1250) — compile-verified
//
#include <hip/hip_runtime.h>
#include <hip/hip_bf16.h>

// Problem constants (match reference)
#define BB      16
#define TT      512
#define C_IN    512
#define C_HID   256
#define C_EMB   384
#define KW      5
#define MAX_MEL 4096   // T * MAX_DUR
#define LRELU   0.3f
#define EPS_LN  1e-5f

typedef float v2f __attribute__((ext_vector_type(2)));
typedef float v8f __attribute__((ext_vector_type(8)));

// ---------------------------------------------------------------------------
// Kernel 1: per-batch cumulative sum of durations (tiny)
// ---------------------------------------------------------------------------
__global__ void cumsum_kernel(const int* __restrict__ dur, int* __restrict__ cum) {
    int b = threadIdx.x;
    if (b < BB) {
        const int* d = dur + b * TT;
        int* c = cum + b * TT;
        int s = 0;
        for (int t = 0; t < TT; ++t) { s += d[t]; c[t] = s; }
    }
}

// ---------------------------------------------------------------------------
// Kernel 1b: repack conv weights (KW, CIN, 256) into WMMA-fragment layout
//   wp[tap][c/4][half][n][v],  half = (c>>1)&1, v = c&1
// so each lane's B fragment is one contiguous float2 (global_load_b64).
// One-time cost, ~4 MB total traffic.
// ---------------------------------------------------------------------------
template<int CIN>
__global__ __launch_bounds__(256) void repack_w_kernel(
    const float* __restrict__ w, float* __restrict__ wp)
{
    const int i = blockIdx.x * 256 + threadIdx.x;   // over KW*CIN*256, n fastest
    const int n    = i & 255;
    const int c    = (i >> 8) % CIN;
    const int tap  = i / (CIN * 256);
    const int chnk = c >> 2;
    const int half = (c >> 1) & 1;
    const int v    = c & 1;
    wp[(((size_t)(tap * (CIN / 4) + chnk) * 2 + half) * 256 + n) * 2 + v] = w[i];
}

// ---------------------------------------------------------------------------
// Kernel 2/3: implicit-GEMM conv1d (K=5, 'same' pad) with f32 WMMA,
// fused bias + leaky-relu + LayerNorm over the 256 output channels.
// Block = 256 threads = 8 waves; block computes 16 rows x 256 channels.
// A tile (20 halo rows x CIN) is staged through LDS (zero-fill once at load,
// divergence-free K-loop, ds_load_2addr_b64, padded stride -> no conflicts).
// B fragments come from the repacked weights: one coalesced b64 per tile.
// ---------------------------------------------------------------------------
template<int CIN>
__global__ __launch_bounds__(256) void conv_ln_kernel(
    const float* __restrict__ in,     // (B*T, CIN)
    const float* __restrict__ wp,     // repacked weights
    const float* __restrict__ bias,   // (256)
    const float* __restrict__ gamma,  // (256)
    const float* __restrict__ beta,   // (256)
    float* __restrict__ out)          // (B*T, 256)
{
    __shared__ __align__(16) float atile[16 + KW - 1][CIN + 4]; // halo'd A tile
    __shared__ float sh[16][264];     // 16 rows x 256 chans (padded stride)
    __shared__ float ps[16][16];
    __shared__ float pq[16][16];
    __shared__ float rmean[16];
    __shared__ float rstd[16];

    const int m0   = blockIdx.x * 16;       // first output row of tile
    const int tid  = threadIdx.x;
    const int wave = tid >> 5;
    const int lane = tid & 31;
    const int half = lane >> 4;
    const int ln   = lane & 15;

    // ---- cooperative, coalesced load of the halo'd input tile ----
    {
        constexpr int C4 = CIN / 4;                 // float4 per row
        const int t0    = m0 & (TT - 1);
        const int bbase = (m0 >> 9) * TT;           // batch * T
        const float4* in4 = (const float4*)in;
        for (int i = tid; i < (16 + KW - 1) * C4; i += 256) {
            const int r    = i / C4;
            const int c4   = i % C4;
            const int tsrc = t0 - 2 + r;
            float4 v = make_float4(0.f, 0.f, 0.f, 0.f);
            if (tsrc >= 0 && tsrc < TT)
                v = in4[(size_t)(bbase + tsrc) * C4 + c4];
            *(float4*)&atile[r][c4 * 4] = v;
        }
    }
    __syncthreads();

    // A-fragment: per ISA 32-bit A layout, lanes 0-15 hold K 0..1 (v0/v1),
    // lanes 16-31 hold K 2..3; M index = lane&15.
    const int koff = half * 2;              // K sub-index within 4-chunk
    const int n0 = wave * 32 + ln;          // B/D column, tile 0
    const int n1 = n0 + 16;                 // tile 1

    v8f acc0 = {};
    v8f acc1 = {};

    for (int tap = 0; tap < KW; ++tap) {
        const float* ar = &atile[ln + tap][koff];   // LDS, no bounds checks
        // packed weight row for this lane: chunk stride = 1024 floats
        const float* wr = wp + (size_t)tap * (CIN / 4) * 1024
                             + ((size_t)half * 256 + n0) * 2;
        #pragma unroll 4
        for (int c4 = 0; c4 < CIN / 4; ++c4) {
            const v2f a  = *(const v2f*)(ar + 4 * c4);
            const v2f b0 = *(const v2f*)(wr + (size_t)c4 * 1024);
            const v2f b1 = *(const v2f*)(wr + (size_t)c4 * 1024 + 32);

            acc0 = __builtin_amdgcn_wmma_f32_16x16x4_f32(
                false, a, false, b0, (short)0, acc0, false, false);
            acc1 = __builtin_amdgcn_wmma_f32_16x16x4_f32(
                false, a, false, b1, (short)0, acc1, false, false);
        }
    }
    __syncthreads();

    // bias + leaky relu -> LDS tile (D layout: row = v + 8*half, col = n)
    #pragma unroll
    for (int v = 0; v < 8; ++v) {
        const int r = v + 8 * half;
        float v0 = acc0[v] + bias[n0];
        v0 = (v0 > 0.f) ? v0 : LRELU * v0;
        sh[r][n0] = v0;
        float v1 = acc1[v] + bias[n1];
        v1 = (v1 > 0.f) ? v1 : LRELU * v1;
        sh[r][n1] = v1;
    }
    __syncthreads();

    // LayerNorm over 256 channels per row (16 rows)
    const int row = tid >> 4;
    const int seg = tid & 15;
    float s = 0.f, q = 0.f;
    #pragma unroll
    for (int i = 0; i < 16; ++i) {
        const float v = sh[row][seg * 16 + i];
        s += v; q += v * v;
    }
    ps[row][seg] = s;
    pq[row][seg] = q;
    __syncthreads();

    if (tid < 16) {
        float S = 0.f, Q = 0.f;
        #pragma unroll
        for (int i = 0; i < 16; ++i) { S += ps[tid][i]; Q += pq[tid][i]; }
        const float mu  = S * (1.f / 256.f);
        const float var = Q * (1.f / 256.f) - mu * mu;
        rmean[tid] = mu;
        rstd[tid]  = rsqrtf(var + EPS_LN);
    }
    __syncthreads();

    const float mu = rmean[row];
    const float rs = rstd[row];
    float* orow = out + (size_t)(m0 + row) * C_HID;
    #pragma unroll
    for (int i = 0; i < 16; ++i) {
        const int c = seg * 16 + i;
        orow[c] = (sh[row][c] - mu) * rs * gamma[c] + beta[c];
    }
}

// ---------------------------------------------------------------------------
// Kernel 4: final linear (256 -> 1) + mask. One wave per row.
// ---------------------------------------------------------------------------
__global__ __launch_bounds__(256) void linear_mask_kernel(
    const float* __restrict__ h,       // (B*T, 256)
    const float* __restrict__ lw,      // (256)
    const float* __restrict__ lb,      // (1)
    const unsigned char* __restrict__ mask, // (B*T) bools
    float* __restrict__ outdur)        // (B*T)
{
    const int row  = blockIdx.x * 8 + (threadIdx.x >> 5);
    const int lane = threadIdx.x & 31;
    const float* hr = h + (size_t)row * C_HID;
    float s = 0.f;
    #pragma unroll
    for (int j = 0; j < 8; ++j)
        s += hr[lane + 32 * j] * lw[lane + 32 * j];
    #pragma unroll
    for (int off = 16; off > 0; off >>= 1)
        s += __shfl_down(s, off, 32);
    if (lane == 0) {
        const float val = s + lb[0];
        outdur[row] = mask[row] ? 0.f : val;
    }
}

// ---------------------------------------------------------------------------
// Kernel 5: length regulation. One block per (b, output-frame p).
// searchsorted-right on cumsum, then float4 row gather for x and emb.
// This stage is pure bandwidth (~470 MB @ 23.3 TB/s -> ~20 us floor).
// ---------------------------------------------------------------------------
__global__ __launch_bounds__(128) void regulate_kernel(
    const float* __restrict__ x,       // (B, T, 512)
    const float* __restrict__ emb,     // (B, T, 384)
    const int*   __restrict__ cum,     // (B, T)
    float* __restrict__ x_up,          // (B, MAX_MEL, 512)
    float* __restrict__ emb_up)        // (B, MAX_MEL, 384)
{
    const int gb = blockIdx.x;
    const int b  = gb >> 12;           // MAX_MEL = 4096
    const int p  = gb & (MAX_MEL - 1);

    __shared__ int s_idx;
    __shared__ int s_valid;
    if (threadIdx.x == 0) {
        const int* c = cum + b * TT;
        const int total = c[TT - 1];
        int lo = 0, hi = TT;
        while (lo < hi) {
            const int mid = (lo + hi) >> 1;
            if (c[mid] <= p) lo = mid + 1; else hi = mid;
        }
        s_idx   = (lo < TT - 1) ? lo : (TT - 1);
        s_valid = (p < total) ? 1 : 0;
    }
    __syncthreads();

    const int idx   = s_idx;
    const int valid = s_valid;
    const float4 z  = make_float4(0.f, 0.f, 0.f, 0.f);
    const int t = threadIdx.x;

    // x row: 512 floats = 128 x float4 -> one per thread
    const float4* xr = (const float4*)(x + ((size_t)b * TT + idx) * C_IN);
    float4* xo = (float4*)(x_up + ((size_t)b * MAX_MEL + p) * C_IN);
    xo[t] = valid ? xr[t] : z;

    // emb row: 384 floats = 96 x float4
    if (t < 96) {
        const float4* er = (const float4*)(emb + ((size_t)b * TT + idx) * C_EMB);
        float4* eo = (float4*)(emb_up + ((size_t)b * MAX_MEL + p) * C_EMB);
        eo[t] = valid ? er[t] : z;
    }
}

// ---------------------------------------------------------------------------
// Launch
// ---------------------------------------------------------------------------
extern "C" void kernel_launch(void* const* d_in, const int* in_sizes, int n_in,
                              void* d_out, int out_size, void* d_ws, size_t ws_size,
                              hipStream_t stream) {
    const float*         x     = (const float*)d_in[0];
    const float*         x_res = (const float*)d_in[1];
    const int*           dur   = (const int*)d_in[2];
    const float*         emb   = (const float*)d_in[3];
    const unsigned char* mask  = (const unsigned char*)d_in[4];
    const float*         c1w   = (const float*)d_in[5];
    const float*         c1b   = (const float*)d_in[6];
    const float*         g1    = (const float*)d_in[7];
    const float*         b1    = (const float*)d_in[8];
    const float*         c2w   = (const float*)d_in[9];
    const float*         c2b   = (const float*)d_in[10];
    const float*         g2    = (const float*)d_in[11];
    const float*         b2    = (const float*)d_in[12];
    const float*         lw    = (const float*)d_in[13];
    const float*         lb    = (const float*)d_in[14];

    float* out    = (float*)d_out;
    float* x_up   = out;                                        // B*MAX_MEL*512
    float* outdur = x_up + (size_t)BB * MAX_MEL * C_IN;         // B*T
    float* emb_up = outdur + (size_t)BB * TT;                   // B*MAX_MEL*384

    float* h1  = (float*)d_ws;                                  // 8192*256 f32
    float* h2  = h1 + (size_t)BB * TT * C_HID;                  // 8192*256 f32
    int*   cum = (int*)(h2 + (size_t)BB * TT * C_HID);          // 16*512 i32
    float* wp1 = (float*)(cum + BB * TT);                       // 5*512*256 f32
    float* wp2 = wp1 + (size_t)KW * C_IN * C_HID;               // 5*256*256 f32

    cumsum_kernel<<<1, BB, 0, stream>>>(dur, cum);

    repack_w_kernel<C_IN> <<<KW * C_IN,  256, 0, stream>>>(c1w, wp1);
    repack_w_kernel<C_HID><<<KW * C_HID, 256, 0, stream>>>(c2w, wp2);

    conv_ln_kernel<C_IN> <<<(BB * TT) / 16, 256, 0, stream>>>(x_res, wp1, c1b, g1, b1, h1);
    conv_ln_kernel<C_HID><<<(BB * TT) / 16, 256, 0, stream>>>(h1,    wp2, c2b, g2, b2, h2);

    linear_mask_kernel<<<(BB * TT) / 8, 256, 0, stream>>>(h2, lw, lb, mask, outdur);

    regulate_kernel<<<BB * MAX_MEL, 128, 0, stream>>>(x, emb, cum, x_up, emb_up);
}